// GraphConvolution_41979010351382
// MI455X (gfx1250) — compile-verified
//
#include <hip/hip_runtime.h>
#include <stdint.h>

typedef __attribute__((ext_vector_type(2))) float v2f;
typedef __attribute__((ext_vector_type(4))) float v4f;
typedef __attribute__((ext_vector_type(8))) float v8f;

#define NF 16          // feature dim
#define CHUNK 256      // edges per half-wave in SpMM

// 16-lane rotate-and-add reduction step via DPP row_ror (wave32: row == 16
// lanes). Rotations 1,2,4,8 give every lane the full 16-lane sum with pure
// VALU ops (no LDS / ds_bpermute traffic).
template <int CTRL>
__device__ __forceinline__ float dpp_ror_add(float v) {
    int r = __builtin_amdgcn_update_dpp(0, __builtin_bit_cast(int, v),
                                        CTRL, 0xF, 0xF, true);
    return v + __builtin_bit_cast(float, r);
}

__device__ __forceinline__ float row16_sum(float p) {
    p = dpp_ror_add<0x121>(p);  // row_ror:1
    p = dpp_ror_add<0x122>(p);  // row_ror:2
    p = dpp_ror_add<0x124>(p);  // row_ror:4
    p = dpp_ror_add<0x128>(p);  // row_ror:8
    return p;
}

// ---------------------------------------------------------------------------
// Kernel 0: zero the hi accumulator (d_ws). Needed every call (atomic accum).
// ---------------------------------------------------------------------------
__global__ void gcn_zero_kernel(float* __restrict__ p, long n) {
    long i = (long)blockIdx.x * blockDim.x + threadIdx.x;
    long stride = (long)gridDim.x * blockDim.x;
    for (; i < n; i += stride) p[i] = 0.0f;
}

// ---------------------------------------------------------------------------
// Kernel 1: COO SpMM, hi = A @ x.  adj_row is sorted.
// Each half-wave (16 lanes = 16 features) processes CHUNK consecutive edges,
// accumulating in a register while the row id is unchanged and flushing with
// one global_atomic_add_f32 per lane on row transitions / chunk boundaries.
// x rows (64B) are read coalesced by the 16 lanes; edge arrays are uniform
// within the half-wave (single cacheline request each, L2-resident x).
// ---------------------------------------------------------------------------
__global__ __launch_bounds__(256) void gcn_spmm_kernel(
    const long long* __restrict__ adj_row,
    const long long* __restrict__ adj_col,
    const float*     __restrict__ adj_val,
    const float*     __restrict__ x,
    float*           __restrict__ hi,
    long E)
{
    long hw = ((long)blockIdx.x * blockDim.x + threadIdx.x) >> 4; // half-wave id
    int  f  = threadIdx.x & 15;                                   // feature lane
    long start = hw * CHUNK;
    if (start >= E) return;
    long end = start + CHUNK;
    if (end > E) end = E;

    float acc = 0.0f;
    long  cur = (long)adj_row[start];
    for (long e = start; e < end; ++e) {
        long r = (long)adj_row[e];
        if (r != cur) {
            unsafeAtomicAdd(&hi[cur * NF + f], acc);  // hw f32 atomic (L2)
            acc = 0.0f;
            cur = r;
        }
        long c = (long)adj_col[e];
        acc = fmaf(adj_val[e], x[c * NF + f], acc);
    }
    unsafeAtomicAdd(&hi[cur * NF + f], acc);
}

// ---------------------------------------------------------------------------
// Kernel 2: fused  out = theta*([hi,h0]@W) + (1-theta)*((1-a)*hi + a*h0) + x
// One wave per 16-node tile. GEMM done as hi@W1 + h0@W2 with 8x
// V_WMMA_F32_16X16X4_F32 accumulating into one 16x16 f32 tile.
// ---------------------------------------------------------------------------
__global__ __launch_bounds__(256) void gcn_fuse_kernel(
    const float* __restrict__ x,
    const float* __restrict__ h0,
    const float* __restrict__ hi,
    const float* __restrict__ weight,   // [2F, F] row-major
    const float* __restrict__ fc_w,     // [1, F]
    const float* __restrict__ fc_b,     // [1]
    const float* __restrict__ lamda,    // [1]
    const int*   __restrict__ lscalar,  // [1]
    float*       __restrict__ out,
    int ntiles)
{
    int wave = (blockIdx.x * blockDim.x + threadIdx.x) >> 5;
    if (wave >= ntiles) return;            // wave-uniform -> EXEC stays all-1s
    int lane = threadIdx.x & 31;
    int half = lane >> 4;                  // 0: rows v..v+7, 1: rows v+8..v+15
    int lp   = lane & 15;                  // column (feature) within tile

    // theta = log(lamda/l + 1): scalar per wave; use v_rcp instead of the
    // full IEEE div_scale/div_fmas sequence (l is a small positive int).
    float theta = logf(fmaf(lamda[0],
                            __builtin_amdgcn_rcpf((float)lscalar[0]), 1.0f));
    float omt   = 1.0f - theta;
    float fcw   = fc_w[lp];
    float fcb   = fc_b[0];

    long base = (long)wave * 16;           // first node of tile

    // ---- C-layout tile loads: VGPR v holds row (v + 8*half), column lp ----
    v8f hi_c, h0_c, x_c;
#pragma unroll
    for (int v = 0; v < 8; ++v) {
        long m = base + v + 8 * half;
        hi_c[v] = hi[m * NF + lp];
        h0_c[v] = h0[m * NF + lp];
        x_c[v]  = x [m * NF + lp];
    }

    // ---- A fragments for the GEMM --------------------------------------
    // A-layout (16x4 f32): lane<16 holds (K=kk, kk+1), lane>=16 (kk+2, kk+3)
    // with kk = 8*(k/2) + 2*(k%2) -> contiguous float2 per step. Two consecutive
    // K-steps are one contiguous float4: load b128 once per step pair.
    const float* hiA = &hi[(base + lp) * NF + 2 * half];
    const float* h0A = &h0[(base + lp) * NF + 2 * half];
    v4f a_hi0 = *(const v4f*)(hiA + 0);    // steps k=0,1
    v4f a_hi1 = *(const v4f*)(hiA + 8);    // steps k=2,3
    v4f a_h00 = *(const v4f*)(h0A + 0);
    v4f a_h01 = *(const v4f*)(h0A + 8);

    // ---- B fragments (weight is 2KB, L2-hot; loaded per wave) ----------
    // B-layout (4x16 f32): V0 = row kk (lanes 0-15) / row kk+2 (lanes 16-31),
    //                      V1 = row kk+1 / kk+3.
    const float* W1 = weight;
    const float* W2 = weight + NF * NF;
    v2f b1[4], b2[4];
#pragma unroll
    for (int k = 0; k < 4; ++k) {
        int kk = 4 * k + 2 * half;
        b1[k][0] = W1[(kk + 0) * NF + lp];
        b1[k][1] = W1[(kk + 1) * NF + lp];
        b2[k][0] = W2[(kk + 0) * NF + lp];
        b2[k][1] = W2[(kk + 1) * NF + lp];
    }

    // ---- GEMM: acc = hi @ W1 + h0 @ W2  (8 x V_WMMA_F32_16X16X4_F32) ----
    v8f acc = {};
    {
        v2f a;
        a[0] = a_hi0[0]; a[1] = a_hi0[1];
        acc = __builtin_amdgcn_wmma_f32_16x16x4_f32(false, a, false, b1[0], (short)0, acc, false, false);
        a[0] = a_hi0[2]; a[1] = a_hi0[3];
        acc = __builtin_amdgcn_wmma_f32_16x16x4_f32(false, a, false, b1[1], (short)0, acc, false, false);
        a[0] = a_hi1[0]; a[1] = a_hi1[1];
        acc = __builtin_amdgcn_wmma_f32_16x16x4_f32(false, a, false, b1[2], (short)0, acc, false, false);
        a[0] = a_hi1[2]; a[1] = a_hi1[3];
        acc = __builtin_amdgcn_wmma_f32_16x16x4_f32(false, a, false, b1[3], (short)0, acc, false, false);
        a[0] = a_h00[0]; a[1] = a_h00[1];
        acc = __builtin_amdgcn_wmma_f32_16x16x4_f32(false, a, false, b2[0], (short)0, acc, false, false);
        a[0] = a_h00[2]; a[1] = a_h00[3];
        acc = __builtin_amdgcn_wmma_f32_16x16x4_f32(false, a, false, b2[1], (short)0, acc, false, false);
        a[0] = a_h01[0]; a[1] = a_h01[1];
        acc = __builtin_amdgcn_wmma_f32_16x16x4_f32(false, a, false, b2[2], (short)0, acc, false, false);
        a[0] = a_h01[2]; a[1] = a_h01[3];
        acc = __builtin_amdgcn_wmma_f32_16x16x4_f32(false, a, false, b2[3], (short)0, acc, false, false);
    }

    // ---- alpha[m] = sum_f hi[m][f]*fc_w[f] + fc_b : DPP row_ror reduce ---
    // (16-lane rotate-add butterfly; result replicated across the 16 column
    //  lanes of each half, matching the C layout.)
    v8f alpha;
#pragma unroll
    for (int v = 0; v < 8; ++v)
        alpha[v] = row16_sum(hi_c[v] * fcw) + fcb;

    // ---- blend + residual, coalesced C-layout store ----------------------
#pragma unroll
    for (int v = 0; v < 8; ++v) {
        long m = base + v + 8 * half;
        float a = alpha[v];
        float r = (1.0f - a) * hi_c[v] + a * h0_c[v];
        out[m * NF + lp] = theta * acc[v] + omt * r + x_c[v];
    }
}

// ---------------------------------------------------------------------------
// Host-side launch
// ---------------------------------------------------------------------------
extern "C" void kernel_launch(void* const* d_in, const int* in_sizes, int n_in,
                              void* d_out, int out_size, void* d_ws, size_t ws_size,
                              hipStream_t stream)
{
    const float*     x       = (const float*)d_in[0];
    const float*     h0      = (const float*)d_in[1];
    const long long* adj_row = (const long long*)d_in[2];  // int64
    const long long* adj_col = (const long long*)d_in[3];  // int64
    const float*     adj_val = (const float*)d_in[4];
    const float*     weight  = (const float*)d_in[5];      // [2F,F]
    const float*     fc_w    = (const float*)d_in[6];
    const float*     fc_b    = (const float*)d_in[7];
    const float*     lamda   = (const float*)d_in[8];
    const int*       lsc     = (const int*)d_in[9];

    long N = (long)in_sizes[0] / NF;       // 500,000
    long E = (long)in_sizes[4];            // 8,000,000
    float* hi  = (float*)d_ws;             // N*F floats scratch
    float* out = (float*)d_out;

    // 0) zero hi
    {
        long n = N * NF;
        gcn_zero_kernel<<<2048, 256, 0, stream>>>(hi, n);
    }
    // 1) SpMM
    {
        long halfwaves = (E + CHUNK - 1) / CHUNK;
        long threads   = halfwaves * 16;
        int  blocks    = (int)((threads + 255) / 256);
        gcn_spmm_kernel<<<blocks, 256, 0, stream>>>(adj_row, adj_col, adj_val,
                                                    x, hi, E);
    }
    // 2) fused gate + GEMM + blend + residual
    {
        int ntiles  = (int)((N + 15) / 16);            // 31250 (N % 16 == 0)
        long threads = (long)ntiles * 32;
        int  blocks  = (int)((threads + 255) / 256);
        gcn_fuse_kernel<<<blocks, 256, 0, stream>>>(x, h0, hi, weight, fc_w,
                                                    fc_b, lamda, lsc, out,
                                                    ntiles);
    }
}